// DifferentiableTransformer_13984413515891
// MI455X (gfx1250) — compile-verified
//
#include <hip/hip_runtime.h>
#include <stdint.h>

// ---------------- static problem configuration (mirrors reference) ----------
#define G0 128
#define G1 128
#define G2 128
#define MAXE 4
#define KPTS 729            // (2*MAXE+1)^3
#define RSTEP 0.02f
#define WAVES_PER_BLOCK 8
#define BLOCK_THREADS (WAVES_PER_BLOCK * 32)

typedef unsigned int v4u __attribute__((ext_vector_type(4)));
typedef int          v8i __attribute__((ext_vector_type(8)));
typedef int          v4i __attribute__((ext_vector_type(4)));

__device__ __forceinline__ float fast_sqrtf(float x) {
#if __has_builtin(__builtin_amdgcn_sqrtf)
    return __builtin_amdgcn_sqrtf(x);   // raw v_sqrt_f32 (<=1 ulp)
#else
    return sqrtf(x);
#endif
}

// ---------------------------------------------------------------------------
// DISASM PROBE (never launched): placed first so the snippet shows exactly
// how __hip_atomic_fetch_add(f32, relaxed, agent) lowers on gfx1250.
// ---------------------------------------------------------------------------
__global__ void atomic_probe_kernel(float* p, float v) {
    __hip_atomic_fetch_add(p + threadIdx.x, v,
                           __ATOMIC_RELAXED, __HIP_MEMORY_SCOPE_AGENT);
}

// ---------------------------------------------------------------------------
// one wave (32 lanes) per atom:
//   1) TDM tensor_load_to_lds stages the atom's R-entry radial row into LDS
//      (async DMA, waited with s_wait_tensorcnt -- per-wave, no barrier)
//   2) row padded with a duplicate of the last entry so the lerp reads are
//      always [ilo], [ilo+1] -> backend can merge into ds_load_2addr_b32
//   3) sweep restructured as 9 z-planes x (<=3 xy-slots per lane): the
//      k%9 / k/9 decode, x/y wrap and dx/dy are hoisted out of the hot loop
// ---------------------------------------------------------------------------
__global__ __launch_bounds__(BLOCK_THREADS)
void dens_splat_kernel(const float* __restrict__ coords,         // (B,N,3)
                       const unsigned char* __restrict__ active, // (B,N) bool
                       const float* __restrict__ occ,            // (B,N)
                       const float* __restrict__ radial,         // (B,N,R)
                       const float* __restrict__ g2c,            // (3,3)
                       float* __restrict__ out,                  // (B,G0,G1,G2)
                       int nAtomsTotal, int N, int R)
{
    extern __shared__ float smem[];   // WAVES_PER_BLOCK * (R+1) floats

    const int wave = threadIdx.x >> 5;   // wave32 (gfx1250)
    const int lane = threadIdx.x & 31;
    const int atom = blockIdx.x * WAVES_PER_BLOCK + wave;
    const bool haveAtom = (atom < nAtomsTotal);
    const int aSafe = haveAtom ? atom : 0;
    const int rowStride = R + 1;         // +1 pad slot

    float* rowLds = &smem[wave * rowStride];
    const float* rowG = radial + (size_t)aSafe * (size_t)R;

    // ---- stage radial row into LDS ----------------------------------------
#if __has_builtin(__builtin_amdgcn_tensor_load_to_lds) && __has_builtin(__builtin_amdgcn_s_wait_tensorcnt)
    if (haveAtom) {
        uint64_t ga = (uint64_t)(uintptr_t)rowG;
        // generic pointer to LDS: low 32 bits are the LDS byte offset
        uint32_t ldsOff = (uint32_t)(uintptr_t)rowLds;

        // D# group 0 (128b): count=1 | lds_addr | global_addr[56:0] | type=2
        v4u g0v;
        g0v.x = 1u;                                   // count=1, user mode
        g0v.y = ldsOff;                               // lds_addr
        g0v.z = (uint32_t)(ga & 0xffffffffull);       // global_addr[31:0]
        g0v.w = (uint32_t)((ga >> 32) & 0x01ffffffull) | (2u << 30); // type=2

        // D# group 1 (256b): 1-D row of R elements, data_size=4B
        v8i g1v;
        g1v[0] = (int)(2u << 16);                 // data_size=2 -> 4 bytes
        g1v[1] = (int)(((uint32_t)R) << 16);      // tensor_dim0[15:0] @ bits 63:48
        g1v[2] = (int)(1u << 16);                 // tensor_dim0[31:16]=0, tensor_dim1=1
        g1v[3] = (int)(((uint32_t)R) << 16);      // tile_dim0 = R
        g1v[4] = 1;                               // tile_dim1 = 1, tile_dim2 = 0
        g1v[5] = R;                               // tensor_dim0_stride[31:0]
        g1v[6] = 0;
        g1v[7] = 0;

        v4i z4 = {0, 0, 0, 0};
#if defined(__clang_major__) && (__clang_major__ >= 23)
        v8i z8 = {0, 0, 0, 0, 0, 0, 0, 0};
        __builtin_amdgcn_tensor_load_to_lds(g0v, g1v, z4, z4, z8, 0);
#else
        __builtin_amdgcn_tensor_load_to_lds(g0v, g1v, z4, z4, 0);
#endif
    }
    __builtin_amdgcn_s_wait_tensorcnt(0);
    asm volatile("" ::: "memory");   // keep LDS reads below the wait
#else
    // cooperative per-wave copy (same-wave producer/consumer, DScnt ordered)
    if (haveAtom) {
        int R4 = R >> 2;
        const float4* s4 = (const float4*)rowG;
        for (int i = lane; i < R4; i += 32) {
            float4 v = s4[i];
            rowLds[4*i+0] = v.x; rowLds[4*i+1] = v.y;
            rowLds[4*i+2] = v.z; rowLds[4*i+3] = v.w;
        }
        for (int i = (R4 << 2) + lane; i < R; i += 32) rowLds[i] = rowG[i];
    }
#endif
    // duplicate last entry into the pad slot: row[R] == row[R-1]
    if (lane == 0) rowLds[R] = rowLds[R - 1];

    // ---- per-atom scalars (wave-uniform) ----------------------------------
    const int b = aSafe / N;
    const float cx = coords[3 * aSafe + 0];
    const float cy = coords[3 * aSafe + 1];
    const float cz = coords[3 * aSafe + 2];
    const float fx = floorf(cx), fy = floorf(cy), fz = floorf(cz);
    const float gx = cx - fx, gy = cy - fy, gz = cz - fz;   // fractional part
    const int cfx = (int)fx, cfy = (int)fy, cfz = (int)fz;
    const float ocv = occ[aSafe] * (active[aSafe] ? 1.0f : 0.0f);

    const float m00 = g2c[0], m01 = g2c[1], m02 = g2c[2];
    const float m10 = g2c[3], m11 = g2c[4], m12 = g2c[5];
    const float m20 = g2c[6], m21 = g2c[7], m22 = g2c[8];

    const uint32_t baseB = (uint32_t)b << 21;   // b * 128^3
    const int Rm1 = R - 1;

    // ---- per-lane xy slots: point p = lane + 32*j of the 9x9 plane --------
    // reference decode for k = iz*81 + p:  ax = p%9-4, ay = (p/9)-4, az = iz-4
    float dxs[3], dys[3];
    uint32_t pxy[3];
    bool     slotOk[3];
#pragma unroll
    for (int j = 0; j < 3; ++j) {
        const int p = lane + 32 * j;
        slotOk[j] = (p < 81);
        const int pp = slotOk[j] ? p : 0;
        const int ax = (pp % 9) - MAXE;
        const int ay = (pp / 9) - MAXE;
        dxs[j] = gx - (float)ax;
        dys[j] = gy - (float)ay;
        pxy[j] = (uint32_t)((cfx + ax) & (G2 - 1))
               | ((uint32_t)((cfy + ay) & (G1 - 1)) << 7);
    }

    // ---- sweep 9 z-planes --------------------------------------------------
    for (int iz = 0; iz < 9; ++iz) {
        const int az = iz - MAXE;
        const float dz = gz - (float)az;
        const float zc0 = m02 * dz;
        const float zc1 = m12 * dz;
        const float zc2 = m22 * dz;
        const uint32_t zTerm = baseB + ((uint32_t)((cfz + az) & (G0 - 1)) << 14);

#pragma unroll
        for (int j = 0; j < 3; ++j) {
            if (!slotOk[j]) continue;   // lane-mask on the partial third slot

            const float c0 = m00 * dxs[j] + m01 * dys[j] + zc0;
            const float c1 = m10 * dxs[j] + m11 * dys[j] + zc1;
            const float c2 = m20 * dxs[j] + m21 * dys[j] + zc2;
            const float dist = fast_sqrtf(c0 * c0 + c1 * c1 + c2 * c2);

            const float rc  = dist * (1.0f / RSTEP);  // 1/0.02f == 50.0f exactly
            const float rcf = floorf(rc);
            const float whi = rc - rcf;               // pre-clamp, as in reference

            int ilo = (int)rcf;
            ilo = ilo < 0 ? 0 : (ilo > Rm1 ? Rm1 : ilo);   // v_med3
            // padded row: row[ilo+1] valid, equals reference row[min(ilo+1,R-1)]
            const float vlo = rowLds[ilo];
            const float vhi = rowLds[ilo + 1];
            const float dens = ((1.0f - whi) * vlo + whi * vhi) * ocv;

            const uint32_t idx = zTerm + pxy[j];
            if (haveAtom) {
                __hip_atomic_fetch_add(&out[idx], dens,
                                       __ATOMIC_RELAXED, __HIP_MEMORY_SCOPE_AGENT);
            }
        }
    }
}

// ---------------------------------------------------------------------------
// zero the output volume (harness poisons d_out with 0xAA, never re-zeroes)
// ---------------------------------------------------------------------------
__global__ void dens_zero_kernel(float4* __restrict__ p4, int n4) {
    int i = blockIdx.x * blockDim.x + threadIdx.x;
    if (i < n4) p4[i] = make_float4(0.f, 0.f, 0.f, 0.f);
}
__global__ void dens_zero_tail_kernel(float* __restrict__ p, int lo, int n) {
    int i = lo + blockIdx.x * blockDim.x + threadIdx.x;
    if (i < n) p[i] = 0.0f;
}

// ---------------------------------------------------------------------------
extern "C" void kernel_launch(void* const* d_in, const int* in_sizes, int n_in,
                              void* d_out, int out_size, void* d_ws, size_t ws_size,
                              hipStream_t stream)
{
    const float*         coords = (const float*)d_in[0];          // (B,N,3)
    const unsigned char* act    = (const unsigned char*)d_in[1];  // (B,N) bool
    const float*         occ    = (const float*)d_in[2];          // (B,N)
    const float*         radial = (const float*)d_in[3];          // (B,N,R)
    const float*         g2c    = (const float*)d_in[4];          // (3,3)
    float*               out    = (float*)d_out;

    const int nAtoms = in_sizes[1];                 // B*N
    const int R      = in_sizes[3] / nAtoms;        // 600
    const int B      = out_size / (G0 * G1 * G2);   // 2
    const int N      = nAtoms / (B > 0 ? B : 1);

    // zero the output volume first (harness poisons it)
    {
        int n4 = out_size >> 2;
        int th = 256;
        int bl = (n4 + th - 1) / th;
        if (bl > 0)
            dens_zero_kernel<<<bl, th, 0, stream>>>((float4*)out, n4);
        int tail = out_size - (n4 << 2);
        if (tail > 0)
            dens_zero_tail_kernel<<<1, 256, 0, stream>>>(out, n4 << 2, out_size);
    }

    // one wave per atom
    {
        int blocks = (nAtoms + WAVES_PER_BLOCK - 1) / WAVES_PER_BLOCK;
        size_t shmem = (size_t)WAVES_PER_BLOCK * (size_t)(R + 1) * sizeof(float);
        dens_splat_kernel<<<blocks, BLOCK_THREADS, shmem, stream>>>(
            coords, act, occ, radial, g2c, out, nAtoms, N, R);
    }
}